// VQLayer_69269232550376
// MI455X (gfx1250) — compile-verified
//
#include <hip/hip_runtime.h>
#include <float.h>

typedef float v2f __attribute__((ext_vector_type(2)));
typedef float v8f __attribute__((ext_vector_type(8)));

#define NUM_EMB   8192
#define EMB_CHUNK 2048   // 32 KB LDS stage
#define N_ROWS    32768  // 2*16*32*32 vectors of dim 4
#define PLANE     16384  // D*H*W
#define N_TILES   2048   // N_ROWS / 16

// One wave (32 lanes) computes the argmin for a 16-row tile of z against all
// 8192 codes using V_WMMA_F32_16X16X4_F32:  D = A(16x4) * B(4x16) + C(16x16)
//   A[m][k]  = z[m][k]                (lane=row, K split across lane halves)
//   B[k][n]  = -2 * emb[n][k]
//   C[m][n]  = ||emb[n]||^2           (per-lane scalar broadcast)
// => D[m][n] = ||emb[n]||^2 - 2 z[m].emb[n]  (argmin-equivalent distance)
__global__ __launch_bounds__(256) void vq_argmin_kernel(
    const float* __restrict__ z, const float4* __restrict__ emb4,
    float* __restrict__ out, float* __restrict__ partial) {
  __shared__ float4 semb[EMB_CHUNK];

  const int lane = threadIdx.x & 31;
  const int wave = threadIdx.x >> 5;
  const int tile = blockIdx.x * 8 + wave;
  const int r    = lane & 15;
  const bool hi  = lane >= 16;

  // ---- A fragment: row m = tile*16 + r; components {0,1} or {2,3} by half
  const int m  = tile * 16 + r;
  const int mb = m >> 14;          // batch index
  const int ms = m & (PLANE - 1);  // spatial index
  const float* zr = z + mb * 4 * PLANE + ms;
  v2f amat;
  amat.x = zr[(hi ? 2 : 0) * PLANE];
  amat.y = zr[(hi ? 3 : 1) * PLANE];

  float best[8];
  int   bidx[8];
#pragma unroll
  for (int v = 0; v < 8; ++v) { best[v] = FLT_MAX; bidx[v] = 0; }

  for (int ch = 0; ch < NUM_EMB / EMB_CHUNK; ++ch) {
    __syncthreads();
    for (int i = threadIdx.x; i < EMB_CHUNK; i += 256)
      semb[i] = emb4[ch * EMB_CHUNK + i];
    __syncthreads();

#pragma unroll 4
    for (int kb = 0; kb < EMB_CHUNK; kb += 16) {
      const int nl = kb + r;               // local code row for this lane
      const float4 e = semb[nl];           // conflict-free: 16 rows x 16B
      const float enorm = e.x * e.x + e.y * e.y + e.z * e.z + e.w * e.w;
      v2f bmat;
      bmat.x = -2.0f * (hi ? e.z : e.x);
      bmat.y = -2.0f * (hi ? e.w : e.y);
      v8f cacc;
#pragma unroll
      for (int v = 0; v < 8; ++v) cacc[v] = enorm;

      v8f dres = __builtin_amdgcn_wmma_f32_16x16x4_f32(
          false, amat, false, bmat, (short)0, cacc, false, false);

      const int n = ch * EMB_CHUNK + nl;   // global code index this lane holds
#pragma unroll
      for (int v = 0; v < 8; ++v) {
        const bool lt = dres[v] < best[v]; // strict < keeps earliest index
        best[v] = lt ? dres[v] : best[v];
        bidx[v] = lt ? n : bidx[v];
      }
    }
  }

  // ---- per-row argmin: reduce across the 16 lanes of each half (N = lane&15)
#pragma unroll
  for (int sft = 1; sft <= 8; sft <<= 1) {
#pragma unroll
    for (int v = 0; v < 8; ++v) {
      const float ov = __shfl_xor(best[v], sft, 32);
      const int   oi = __shfl_xor(bidx[v], sft, 32);
      const bool take = (ov < best[v]) || ((ov == best[v]) && (oi < bidx[v]));
      best[v] = take ? ov : best[v];
      bidx[v] = take ? oi : bidx[v];
    }
  }

  // ---- writers: lane v -> row v (half0), lane 16+v -> row 8+v (half1)
  float lacc = 0.0f;
#pragma unroll
  for (int v = 0; v < 8; ++v) {
    if (r == v) {
      const int row = tile * 16 + v + (hi ? 8 : 0);
      const float4 e = emb4[bidx[v]];
      const int b = row >> 14;
      const int s = row & (PLANE - 1);
      float* op = out + b * 4 * PLANE + s;
      const float* zp = z + b * 4 * PLANE + s;
      const float d0 = e.x - zp[0 * PLANE];
      const float d1 = e.y - zp[1 * PLANE];
      const float d2 = e.z - zp[2 * PLANE];
      const float d3 = e.w - zp[3 * PLANE];
      op[0 * PLANE] = e.x;  // straight-through forward value == emb[idx]
      op[1 * PLANE] = e.y;
      op[2 * PLANE] = e.z;
      op[3 * PLANE] = e.w;
      lacc = d0 * d0 + d1 * d1 + d2 * d2 + d3 * d3;
    }
  }
  // deterministic fixed-order wave sum of the 16 row residuals
#pragma unroll
  for (int sft = 1; sft <= 16; sft <<= 1)
    lacc += __shfl_xor(lacc, sft, 32);
  if (lane == 0) partial[tile] = lacc;
}

// Deterministic single-block reduction of 2048 per-tile partials -> loss.
__global__ __launch_bounds__(256) void vq_loss_reduce(
    const float* __restrict__ partial, float* __restrict__ out) {
  __shared__ float red[256];
  const int tid = threadIdx.x;
  float s = 0.0f;
  for (int i = tid; i < N_TILES; i += 256) s += partial[i];
  red[tid] = s;
  __syncthreads();
  for (int off = 128; off > 0; off >>= 1) {
    if (tid < off) red[tid] += red[tid + off];
    __syncthreads();
  }
  if (tid == 0) {
    // loss = mean((zq-z)^2) + 0.25*mean((zq-z)^2) = 1.25 * sum / (N*4)
    out[4 * N_ROWS] = 1.25f * red[0] / (float)(4 * N_ROWS);
  }
}

extern "C" void kernel_launch(void* const* d_in, const int* in_sizes, int n_in,
                              void* d_out, int out_size, void* d_ws, size_t ws_size,
                              hipStream_t stream) {
  const float* z   = (const float*)d_in[0];   // [2,4,16,32,32] f32
  const float* emb = (const float*)d_in[1];   // [8192,4] f32
  float* out       = (float*)d_out;           // 131072 z_q floats + 1 loss
  float* partial   = (float*)d_ws;            // 2048 floats scratch

  vq_argmin_kernel<<<256, 256, 0, stream>>>(z, (const float4*)emb, out, partial);
  vq_loss_reduce<<<1, 256, 0, stream>>>(partial, out);
}